// Kvmemnn_326417514921
// MI455X (gfx1250) — compile-verified
//
#include <hip/hip_runtime.h>
#include <hip/hip_bf16.h>

typedef __attribute__((ext_vector_type(2))) float v2f;
typedef __attribute__((ext_vector_type(8))) float v8f;

#define EMB_D      128
#define MAX_NORM   10.0f
#define RENORM_EPS 1e-7f
#define COS_EPS    1e-8f
#define N_MEM      1000
#define N_MEMROWS  1008   // padded to 63 WMMA tiles of 16
#define N_CAND     100
#define OUT_ROWS   101

// ---------------------------------------------------------------------------
// Stage 1: one wave per sequence. tf-idf weighted bag of embeddings with
// torch-style max_norm renorm at lookup. Fully coalesced: each lane does a
// float4 (b128) load so the wave fetches exactly one 512B embedding row.
// Row norms computed with wave shuffle reductions (wave32).
// ---------------------------------------------------------------------------
__global__ __launch_bounds__(256) void
kvmemnn_encode(const int* __restrict__ xs, const int* __restrict__ mems,
               const int* __restrict__ ys, const int* __restrict__ cands,
               const float* __restrict__ lt, const float* __restrict__ freqs,
               float* __restrict__ enc,   // [1008][128] workspace
               float* __restrict__ norms, // [1008]      workspace
               float* __restrict__ out2)  // d_out + 101*128 (ys/cands encs)
{
    const int gw   = (blockIdx.x * blockDim.x + threadIdx.x) >> 5; // global wave
    const int lane = threadIdx.x & 31;

    const int* idxp; int L; float* dst;
    bool wantNorm = false; int normIdx = 0;

    if (gw < N_MEM) {                       // memory sequences
        idxp = mems + gw * 64; L = 64; dst = enc + gw * EMB_D;
        wantNorm = true; normIdx = gw;
    } else if (gw == N_MEM) {               // query xs -> memory row 1000
        idxp = xs; L = 64; dst = enc + N_MEM * EMB_D;
        wantNorm = true; normIdx = N_MEM;
    } else if (gw < N_MEMROWS) {            // zero pad rows for WMMA tiling
        float4 z = {0.f, 0.f, 0.f, 0.f};
        *(float4*)(enc + gw * EMB_D + lane * 4) = z;
        if (lane == 0) norms[gw] = 0.f;
        return;
    } else if (gw == N_MEMROWS) {           // ys -> out2 row 0
        idxp = ys; L = 32; dst = out2;
    } else if (gw < N_MEMROWS + 1 + N_CAND) { // candidates -> out2 rows 1..100
        idxp = cands + (gw - N_MEMROWS - 1) * 32; L = 32;
        dst = out2 + (gw - N_MEMROWS) * EMB_D;
    } else {
        return;
    }

    // token indices + tf-idf weights for this wave's sequence
    const int   i0 = idxp[lane];
    const int   i1 = (L == 64) ? idxp[lane + 32] : 0;
    const float w0 = freqs[i0];
    const float w1 = (L == 64) ? freqs[i1] : 0.f;

    float ws2 = w0 * w0 + w1 * w1;          // ||w||^2 via wave reduction
    #pragma unroll
    for (int m = 16; m >= 1; m >>= 1) ws2 += __shfl_xor(ws2, m, 32);
    const float invw = rsqrtf(ws2);

    float4 acc = {0.f, 0.f, 0.f, 0.f};
    for (int l = 0; l < L; ++l) {
        const int   idx = __shfl((l < 32) ? i0 : i1, l & 31, 32);
        const float wl  = __shfl((l < 32) ? w0 : w1, l & 31, 32) * invw;
        const float4 e  = *(const float4*)(lt + (size_t)idx * EMB_D + lane * 4);
        // per-row L2 norm (wave reduction), torch max_norm renorm
        float rs = e.x * e.x + e.y * e.y + e.z * e.z + e.w * e.w;
        #pragma unroll
        for (int m = 16; m >= 1; m >>= 1) rs += __shfl_xor(rs, m, 32);
        const float rn    = sqrtf(rs);
        const float scale = (rn > MAX_NORM) ? (MAX_NORM / (rn + RENORM_EPS)) : 1.0f;
        const float c     = wl * scale;
        acc.x += c * e.x; acc.y += c * e.y; acc.z += c * e.z; acc.w += c * e.w;
    }
    *(float4*)(dst + lane * 4) = acc;

    if (wantNorm) {                         // encoding L2 norm for cosine sim
        float a2 = acc.x * acc.x + acc.y * acc.y + acc.z * acc.z + acc.w * acc.w;
        #pragma unroll
        for (int m = 16; m >= 1; m >>= 1) a2 += __shfl_xor(a2, m, 32);
        if (lane == 0) norms[normIdx] = sqrtf(a2);
    }
}

// ---------------------------------------------------------------------------
// Stage 2: single workgroup. Cosine sims via V_WMMA_F32_16X16X4_F32
// (16 memory rows per tile, query vector replicated across all 16 B-columns),
// block softmax over 1001, attention-weighted sum, broadcast write.
// ---------------------------------------------------------------------------
__global__ __launch_bounds__(256) void
kvmemnn_attend(const float* __restrict__ enc, const float* __restrict__ norms,
               float* __restrict__ out)
{
    __shared__ float xsv[EMB_D];
    __shared__ float sims[N_MEMROWS];
    __shared__ float prob[N_MEMROWS];
    __shared__ float red[256];

    const int t    = threadIdx.x;
    const int lane = t & 31;
    const int wave = t >> 5;

    if (t < EMB_D) xsv[t] = enc[N_MEM * EMB_D + t];
    __syncthreads();

    const float nx = fmaxf(norms[N_MEM], COS_EPS);

    // --- WMMA cosine similarities: 63 tiles of 16 rows, K=128 in steps of 4.
    // A (16x4 f32, 2 VGPRs): lanes 0-15 K=k0,k0+1 ; lanes 16-31 K=k0+2,k0+3.
    // B (4x16): query chunk replicated across all 16 columns.
    for (int tile = wave; tile < N_MEMROWS / 16; tile += 8) {
        const int row  = tile * 16 + (lane & 15);
        const int koff = (lane >> 4) * 2;
        const float* ar = enc + (size_t)row * EMB_D;

        v8f c = {};
        for (int k0 = 0; k0 < EMB_D; k0 += 4) {
            v2f a, b;
            a.x = ar[k0 + koff];      a.y = ar[k0 + koff + 1];
            b.x = xsv[k0 + koff];     b.y = xsv[k0 + koff + 1];
            c = __builtin_amdgcn_wmma_f32_16x16x4_f32(
                    false, a, false, b, (short)0, c, false, false);
        }
        // D layout: element (m, n=0) lives in VGPR m&7, lane (m>=8 ? 16 : 0)
        float mysim = 0.f;
        #pragma unroll
        for (int m = 0; m < 16; ++m) {
            const float v = __shfl(c[m & 7], (m >= 8) ? 16 : 0, 32);
            if (lane == m) mysim = v;
        }
        if (lane < 16) {
            const int r  = tile * 16 + lane;
            const float nm = fmaxf(norms[r], COS_EPS);
            sims[r] = mysim / (nx * nm);
        }
    }
    __syncthreads();

    // --- softmax over the 1001 valid rows ---
    float lm = -3.4e38f;
    for (int m = t; m < N_MEM + 1; m += 256) lm = fmaxf(lm, sims[m]);
    red[t] = lm; __syncthreads();
    #pragma unroll
    for (int s = 128; s > 0; s >>= 1) {
        if (t < s) red[t] = fmaxf(red[t], red[t + s]);
        __syncthreads();
    }
    const float mx = red[0];
    __syncthreads();

    float ls = 0.f;
    for (int m = t; m < N_MEM + 1; m += 256) {
        const float e = expf(sims[m] - mx);
        prob[m] = e; ls += e;
    }
    red[t] = ls; __syncthreads();
    #pragma unroll
    for (int s = 128; s > 0; s >>= 1) {
        if (t < s) red[t] += red[t + s];
        __syncthreads();
    }
    const float inv = 1.0f / red[0];

    // --- lhs = attn @ mems_enc ; broadcast to the 101 output rows ---
    if (t < EMB_D) {
        float acc = 0.f;
        for (int m = 0; m < N_MEM + 1; ++m)
            acc += prob[m] * enc[(size_t)m * EMB_D + t];
        const float lhs = acc * inv;
        for (int r = 0; r < OUT_ROWS; ++r) out[r * EMB_D + t] = lhs;
    }
}

// ---------------------------------------------------------------------------
extern "C" void kernel_launch(void* const* d_in, const int* in_sizes, int n_in,
                              void* d_out, int out_size, void* d_ws, size_t ws_size,
                              hipStream_t stream)
{
    (void)in_sizes; (void)n_in; (void)out_size; (void)ws_size;
    const int*   xs    = (const int*)d_in[0];   // [1,64]
    const int*   mems  = (const int*)d_in[1];   // [1000,64]
    const int*   ys    = (const int*)d_in[2];   // [1,32]
    const int*   cands = (const int*)d_in[3];   // [100,32]
    const float* lt    = (const float*)d_in[4]; // [500000,128]
    const float* freqs = (const float*)d_in[5]; // [500000]

    float* out   = (float*)d_out;               // [101*128] + [101*128]
    float* enc   = (float*)d_ws;                // [1008][128]
    float* norms = enc + N_MEMROWS * EMB_D;     // [1008]
    float* out2  = out + OUT_ROWS * EMB_D;      // ys/cands encodings

    // 1109 working waves (+pad), 8 waves per 256-thread block
    const int nwaves = N_MEMROWS + 1 + N_CAND;  // 1109
    const int blocks = (nwaves + 7) / 8;
    kvmemnn_encode<<<blocks, 256, 0, stream>>>(xs, mems, ys, cands, lt, freqs,
                                               enc, norms, out2);
    kvmemnn_attend<<<1, 256, 0, stream>>>(enc, norms, out);
}